// MultiplicativeAttention_11699490914913
// MI455X (gfx1250) — compile-verified
//
#include <hip/hip_runtime.h>

// Problem constants (fixed by the reference harness).
#define BB 32
#define SS 4096
#define DD 1024
#define NEGV (-1e9f)

typedef __attribute__((ext_vector_type(2))) float v2f;
typedef __attribute__((ext_vector_type(8))) float v8f;

// ---------------------------------------------------------------------------
// Kernel 1: q = query @ W_q^T using V_WMMA_F32_16X16X4_F32.
// One wave per 16x16 output tile. A[m,k] = query[m0+m, k0+k],
// B[k,n] = W_q[n0+n, k0+k]  (so D = q tile).
// f32 A 16x4 lane layout: lanes 0-15 -> M=lane, K={0,1}; lanes 16-31 -> K={2,3}.
// B 4x16 mirrors it with lane = N. Both reduce to per-lane float2 loads at
// row*D + k0 + (hi?2:0).
// C/D layout: VGPR r, lane l -> (M = r + (l>=16 ? 8:0), N = l&15).
// ---------------------------------------------------------------------------
__global__ __launch_bounds__(32) void qproj_wmma(const float* __restrict__ query,
                                                 const float* __restrict__ Wq,
                                                 float* __restrict__ qout) {
  const int lane = threadIdx.x & 31;
  const int lm   = lane & 15;
  const bool hi  = lane >= 16;
  const int n0   = blockIdx.x * 16;   // output feature tile (0..1023)
  const int m0   = blockIdx.y * 16;   // batch-row tile (0..31)
  const int koff = hi ? 2 : 0;

  const float* arow = query + (size_t)(m0 + lm) * DD;
  const float* brow = Wq    + (size_t)(n0 + lm) * DD;

  v8f acc = {0.f, 0.f, 0.f, 0.f, 0.f, 0.f, 0.f, 0.f};

#pragma unroll 4
  for (int k0 = 0; k0 < DD; k0 += 4) {
    v2f a = *(const v2f*)(arow + k0 + koff);   // 8B aligned (k0+koff even)
    v2f b = *(const v2f*)(brow + k0 + koff);
    acc = __builtin_amdgcn_wmma_f32_16x16x4_f32(
        /*neg_a=*/false, a, /*neg_b=*/false, b,
        /*c_mod=*/(short)0, acc, /*reuse_a=*/false, /*reuse_b=*/false);
  }

  float* dst = qout + (size_t)(m0 + (hi ? 8 : 0)) * DD + n0 + lm;
#pragma unroll
  for (int r = 0; r < 8; ++r) dst[(size_t)r * DD] = acc[r];
}

// ---------------------------------------------------------------------------
// Kernel 2: energies[b,s] = dot(keys[b,s,:], q[b,:]) (masked).
// One wave per s-row; q[b,:] staged in LDS; float4 coalesced key loads.
// Bandwidth-bound: streams keys (512 MB) exactly once.
// ---------------------------------------------------------------------------
__global__ __launch_bounds__(256) void energies_kernel(const float* __restrict__ keys,
                                                       const int* __restrict__ mask,
                                                       const float* __restrict__ q,
                                                       float* __restrict__ energ) {
  __shared__ float qs[DD];
  const int b   = blockIdx.y;
  const int tid = threadIdx.x;
#pragma unroll
  for (int i = 0; i < DD / 256; ++i)
    qs[tid + i * 256] = q[(size_t)b * DD + tid + i * 256];
  __syncthreads();

  const int lane = tid & 31;
  const int wave = tid >> 5;
  const int s    = blockIdx.x * 8 + wave;
  const float* krow = keys + ((size_t)b * SS + s) * DD;

  float acc = 0.f;
#pragma unroll
  for (int i = 0; i < 8; ++i) {
    const int d = i * 128 + lane * 4;          // 512B coalesced per step
    const float4 kv = *(const float4*)(krow + d);
    acc += kv.x * qs[d] + kv.y * qs[d + 1] + kv.z * qs[d + 2] + kv.w * qs[d + 3];
  }
  // wave32 butterfly reduction
  for (int off = 16; off > 0; off >>= 1) acc += __shfl_xor(acc, off, 32);

  if (lane == 0) {
    const float e = (mask[(size_t)b * SS + s] == 0) ? NEGV : acc;
    energ[(size_t)b * SS + s] = e;
  }
}

// ---------------------------------------------------------------------------
// Kernel 3: per-batch softmax stats (max, sum-of-exp) over S=4096.
// ---------------------------------------------------------------------------
__global__ __launch_bounds__(256) void softmax_stats(const float* __restrict__ energ,
                                                     float* __restrict__ stats) {
  __shared__ float red[256];
  const int b   = blockIdx.x;
  const int tid = threadIdx.x;
  const float* e = energ + (size_t)b * SS;

  float m = -__builtin_inff();
  for (int i = tid; i < SS; i += 256) m = fmaxf(m, e[i]);
  red[tid] = m;
  __syncthreads();
  for (int off = 128; off > 0; off >>= 1) {
    if (tid < off) red[tid] = fmaxf(red[tid], red[tid + off]);
    __syncthreads();
  }
  const float mx = red[0];
  __syncthreads();

  float sm = 0.f;
  for (int i = tid; i < SS; i += 256) sm += expf(e[i] - mx);
  red[tid] = sm;
  __syncthreads();
  for (int off = 128; off > 0; off >>= 1) {
    if (tid < off) red[tid] += red[tid + off];
    __syncthreads();
  }
  if (tid == 0) {
    stats[2 * b]     = mx;
    stats[2 * b + 1] = red[0];
  }
}

// ---------------------------------------------------------------------------
// Kernel 4: attn[b,s] = exp(e - max) / sum   (written to d_out attn region)
// ---------------------------------------------------------------------------
__global__ __launch_bounds__(256) void attn_norm(const float* __restrict__ energ,
                                                 const float* __restrict__ stats,
                                                 float* __restrict__ attn) {
  const int idx = blockIdx.x * 256 + threadIdx.x;
  const int b   = idx >> 12;  // S = 4096
  const float inv = 1.0f / stats[2 * b + 1];
  attn[idx] = expf(energ[idx] - stats[2 * b]) * inv;
}

// ---------------------------------------------------------------------------
// Kernel 5: context[b,v] = sum_s attn[b,s] * values[b,s,v].
// Block = (b, 256-wide v chunk); attn chunk staged in LDS; each s-iteration
// reads a fully coalesced 1KB slice of a values row. Streams values (512 MB)
// exactly once. Prefetch hint -> global_prefetch_b8.
// ---------------------------------------------------------------------------
__global__ __launch_bounds__(256) void context_kernel(const float* __restrict__ values,
                                                      const float* __restrict__ attn,
                                                      float* __restrict__ ctx) {
  __shared__ float aw[256];
  const int b   = blockIdx.y;
  const int v   = blockIdx.x * 256 + threadIdx.x;
  const float* vb = values + ((size_t)b * SS) * DD + v;

  float acc = 0.f;
  for (int sc = 0; sc < SS / 256; ++sc) {
    aw[threadIdx.x] = attn[(size_t)b * SS + sc * 256 + threadIdx.x];
    __syncthreads();
    if (sc + 1 < SS / 256)
      __builtin_prefetch(vb + (size_t)(sc + 1) * 256 * DD, 0, 1);
#pragma unroll 8
    for (int j = 0; j < 256; ++j) {
      acc += aw[j] * vb[(size_t)(sc * 256 + j) * DD];
    }
    __syncthreads();
  }
  ctx[(size_t)b * DD + v] = acc;
}

// ---------------------------------------------------------------------------
extern "C" void kernel_launch(void* const* d_in, const int* in_sizes, int n_in,
                              void* d_out, int out_size, void* d_ws, size_t ws_size,
                              hipStream_t stream) {
  (void)in_sizes; (void)n_in; (void)out_size; (void)ws_size;

  const float* query  = (const float*)d_in[0];   // (B, D)
  const float* keys   = (const float*)d_in[1];   // (B, S, D)
  const float* values = (const float*)d_in[2];   // (B, S, D)
  const int*   mask   = (const int*)d_in[3];     // (B, S)
  const float* Wq     = (const float*)d_in[4];   // (D, D)

  float* ctx  = (float*)d_out;        // (B, D)   = 32768 floats
  float* attn = ctx + BB * DD;        // (B, S)   = 131072 floats

  float* ws    = (float*)d_ws;
  float* qws   = ws;                  // (B, D)   32768 floats
  float* energ = qws + BB * DD;       // (B, S)   131072 floats
  float* stats = energ + BB * SS;     // (B, 2)   64 floats

  qproj_wmma<<<dim3(DD / 16, BB / 16), 32, 0, stream>>>(query, Wq, qws);
  energies_kernel<<<dim3(SS / 8, BB), 256, 0, stream>>>(keys, mask, qws, energ);
  softmax_stats<<<BB, 256, 0, stream>>>(energ, stats);
  attn_norm<<<(BB * SS) / 256, 256, 0, stream>>>(energ, stats, attn);
  context_kernel<<<dim3(DD / 256, BB), 256, 0, stream>>>(values, attn, ctx);
}